// NeuronFFT_80427557585263
// MI455X (gfx1250) — compile-verified
//
#include <hip/hip_runtime.h>
#include <math.h>

// ---------------------------------------------------------------------------
// Batched 4096-point complex FFT, radix-16 (3 stages), DFT-16 via f32 WMMA.
//   B = 8192 rows, N = 4096. One workgroup (256 thr = 8 waves) per row.
//   Bandwidth-bound: 512 MiB total traffic -> ~22us floor @ 23.3 TB/s, so the
//   global side uses b128 accesses; all arithmetic rides the WMMA pipe.
//   Each wave owns 2 tiles of 16 DFT columns per stage; DFT-16 matmul done
//   with chained V_WMMA_F32_16X16X4_F32 (K=16 -> 4 chained WMMAs per real
//   matmul, 16 WMMAs per complex 16x16x16 tile; 96 WMMA total per wave).
// ---------------------------------------------------------------------------

typedef __attribute__((ext_vector_type(2))) float v2f;
typedef __attribute__((ext_vector_type(8))) float v8f;

#define N_FFT   4096
#define THREADS 256
#define PI_F    3.14159265358979323846f

// LDS padding: +1 float every 16 to break stride-16 / stride-256 conflicts.
__device__ __forceinline__ int ldspad(int a) { return a + (a >> 4); }
#define LDS_ELEMS (N_FFT + (N_FFT >> 4))   // 4352 floats per component

// base-16 digit reversal of a 12-bit index: d2 d1 d0 -> d0 d1 d2
__device__ __forceinline__ int digit_rev16(int i) {
    return ((i & 0x00F) << 8) | (i & 0x0F0) | ((i >> 8) & 0x00F);
}

__global__ __launch_bounds__(THREADS)
void fft4096_radix16_wmma(const float* __restrict__ xr,
                          const float* __restrict__ xi,
                          float* __restrict__ yr,
                          float* __restrict__ yi)
{
    __shared__ float lr[LDS_ELEMS];
    __shared__ float li[LDS_ELEMS];

    const int  tid  = threadIdx.x;
    const long base = (long)blockIdx.x * N_FFT;

    // ---- load: b128 coalesced global reads, digit-reversed LDS scatter ----
    // Thread handles 4 chunks of 4 consecutive elements. Within a 4-aligned
    // chunk only digit d0 varies, so drev(i+k) = drev(i) + 256*k.
    #pragma unroll
    for (int it = 0; it < N_FFT / (THREADS * 4); ++it) {
        const int i = (tid + it * THREADS) * 4;
        __builtin_prefetch(&xr[base + ((i + 2048) & (N_FFT - 1))], 0, 0);
        __builtin_prefetch(&xi[base + ((i + 2048) & (N_FFT - 1))], 0, 0);
        const float4 r4 = *reinterpret_cast<const float4*>(&xr[base + i]);
        const float4 i4 = *reinterpret_cast<const float4*>(&xi[base + i]);
        const int rev = digit_rev16(i);
        lr[ldspad(rev)]       = r4.x;  li[ldspad(rev)]       = i4.x;
        lr[ldspad(rev + 256)] = r4.y;  li[ldspad(rev + 256)] = i4.y;
        lr[ldspad(rev + 512)] = r4.z;  li[ldspad(rev + 512)] = i4.z;
        lr[ldspad(rev + 768)] = r4.w;  li[ldspad(rev + 768)] = i4.w;
    }
    __syncthreads();

    const int lane = tid & 31;
    const int wave = tid >> 5;
    const int nidx = lane & 15;   // N index for B/C/D fragments; M for A
    const int hi   = lane >> 4;   // which lane-half

    // ---- constant DFT-16 matrix fragments (A operand), built once ----
    // A is 16x16 split into four 16x4 K-blocks. f32 A layout:
    //   lane = M (0..15, replicated across halves), K = 2*hi + v  (v = VGPR)
    v2f Ar[4], Ai[4];
    #pragma unroll
    for (int kb = 0; kb < 4; ++kb) {
        #pragma unroll
        for (int v = 0; v < 2; ++v) {
            const int k = 4 * kb + 2 * hi + v;
            float s, c;
            __sincosf(-2.0f * PI_F * (float)(nidx * k) / 16.0f, &s, &c);
            Ar[kb][v] = c;
            Ai[kb][v] = s;
        }
    }

    // ---- 3 radix-16 stages: L = 1, 16, 256 ----
    #pragma unroll
    for (int s = 0; s < 3; ++s) {
        const int L = 1 << (4 * s);

        // twiddle pass: element e = g*16L + j*L + q gets * W_{16L}^{j*q}
        if (s > 0) {
            const float wb = -2.0f * PI_F / (float)(16 * L);
            #pragma unroll
            for (int it = 0; it < N_FFT / THREADS; ++it) {
                const int e = tid + it * THREADS;
                const int j = (e >> (4 * s)) & 15;
                const int q = e & (L - 1);
                if (j * q != 0) {
                    float sn, cs;
                    __sincosf(wb * (float)(j * q), &sn, &cs);
                    const int   p = ldspad(e);
                    const float r = lr[p], im = li[p];
                    lr[p] = r * cs - im * sn;
                    li[p] = r * sn + im * cs;
                }
            }
            __syncthreads();
        }

        // matmul pass: 256 DFT columns -> 16 tiles of 16; wave w owns 2 tiles.
        #pragma unroll
        for (int tt = 0; tt < 2; ++tt) {
            const int tile  = wave * 2 + tt;
            const int col   = tile * 16 + nidx;       // this lane's column
            const int g     = col / L;                // group
            const int q     = col - g * L;            // offset in group
            const int cbase = g * 16 * L + q;         // elem(j) = cbase + j*L

            // gather B fragments (16(K=j) x 16(N=col) data tile) from LDS
            v2f Br[4], Bi[4];
            #pragma unroll
            for (int kb = 0; kb < 4; ++kb) {
                #pragma unroll
                for (int v = 0; v < 2; ++v) {
                    const int j = 4 * kb + 2 * hi + v;
                    const int p = ldspad(cbase + j * L);
                    Br[kb][v] = lr[p];
                    Bi[kb][v] = li[p];
                }
            }

            // complex 16x16x16 matmul via chained f32 WMMA (K=4 each):
            //   Dr = Ar*Br - Ai*Bi      Di = Ar*Bi + Ai*Br
            v8f accRR = {}, accII = {}, accI = {};
            #pragma unroll
            for (int kb = 0; kb < 4; ++kb)
                accRR = __builtin_amdgcn_wmma_f32_16x16x4_f32(
                    false, Ar[kb], false, Br[kb], (short)0, accRR, false, false);
            #pragma unroll
            for (int kb = 0; kb < 4; ++kb)
                accII = __builtin_amdgcn_wmma_f32_16x16x4_f32(
                    false, Ai[kb], false, Bi[kb], (short)0, accII, false, false);
            #pragma unroll
            for (int kb = 0; kb < 4; ++kb)
                accI = __builtin_amdgcn_wmma_f32_16x16x4_f32(
                    false, Ar[kb], false, Bi[kb], (short)0, accI, false, false);
            #pragma unroll
            for (int kb = 0; kb < 4; ++kb)
                accI = __builtin_amdgcn_wmma_f32_16x16x4_f32(
                    false, Ai[kb], false, Br[kb], (short)0, accI, false, false);

            // scatter D back: lane holds rows M = v + 8*hi of column `col`
            #pragma unroll
            for (int v = 0; v < 8; ++v) {
                const int r = v + 8 * hi;
                const int p = ldspad(cbase + r * L);
                lr[p] = accRR[v] - accII[v];
                li[p] = accI[v];
            }
        }
        __syncthreads();
    }

    // ---- store: natural order, b128 coalesced global writes ----
    // Blocks of 4 (4-aligned) never cross a 16-block pad boundary, so the
    // 4 LDS reads are contiguous; the global store is a single b128.
    #pragma unroll
    for (int it = 0; it < N_FFT / (THREADS * 4); ++it) {
        const int i = (tid + it * THREADS) * 4;
        const int p = ldspad(i);              // pad(i+k) = pad(i)+k, k=0..3
        float4 r4, i4;
        r4.x = lr[p];     i4.x = li[p];
        r4.y = lr[p + 1]; i4.y = li[p + 1];
        r4.z = lr[p + 2]; i4.z = li[p + 2];
        r4.w = lr[p + 3]; i4.w = li[p + 3];
        *reinterpret_cast<float4*>(&yr[base + i]) = r4;
        *reinterpret_cast<float4*>(&yi[base + i]) = i4;
    }
}

extern "C" void kernel_launch(void* const* d_in, const int* in_sizes, int n_in,
                              void* d_out, int out_size, void* d_ws, size_t ws_size,
                              hipStream_t stream)
{
    (void)n_in; (void)out_size; (void)d_ws; (void)ws_size;
    const float* xr = (const float*)d_in[0];
    const float* xi = (const float*)d_in[1];
    const int total = in_sizes[0];          // B * N
    const int rows  = total / N_FFT;        // 8192
    float* yr = (float*)d_out;              // first output of the tuple
    float* yi = yr + (size_t)total;         // second output, concatenated

    fft4096_radix16_wmma<<<dim3(rows), dim3(THREADS), 0, stream>>>(xr, xi, yr, yi);
}